// SO3Unpooling_50345606643915
// MI455X (gfx1250) — compile-verified
//
#include <hip/hip_runtime.h>

typedef float v2f __attribute__((ext_vector_type(2)));
typedef float v8f __attribute__((ext_vector_type(8)));

#define PI_D 3.14159265358979323846
#define LSTR 65   // 64 + 1 padding (bank-conflict free column reads)

__device__ __forceinline__ int off_l(int l) { return l * (4 * l * l - 1) / 3; }

// ---------------------------------------------------------------------------
// Wigner small-d via explicit Jacobi sum (double precision, lgamma form).
// beta in (0, pi) so cos(beta/2), sin(beta/2) in (0,1): logs are finite.
// ---------------------------------------------------------------------------
__device__ double wig_d(int l, int mp, int m, double beta) {
  double cb = cos(0.5 * beta), sb = sin(0.5 * beta);
  double lcb = log(cb), lsb = log(sb);
  double pref = 0.5 * (lgamma((double)(l + mp + 1)) + lgamma((double)(l - mp + 1)) +
                       lgamma((double)(l + m + 1)) + lgamma((double)(l - m + 1)));
  int smin = m - mp; if (smin < 0) smin = 0;
  int smax = l + m;  if (l - mp < smax) smax = l - mp;
  double val = 0.0;
  for (int s = smin; s <= smax; ++s) {
    double e = pref - lgamma((double)(l + m - s + 1)) - lgamma((double)(s + 1))
                    - lgamma((double)(mp - m + s + 1)) - lgamma((double)(l - mp - s + 1))
             + (double)(2 * l + m - mp - 2 * s) * lcb
             + (double)(mp - m + 2 * s) * lsb;
    double term = exp(e);
    val += ((mp - m + s) & 1) ? -term : term;
  }
  return val;
}

// SOFT quadrature weights for b=16 (32 betas)
__global__ void k_weights16(float* __restrict__ w16) {
  int j = threadIdx.x;                          // 0..31
  double s = 0.0;
  for (int k = 0; k < 16; ++k)
    s += sin((2.0 * j + 1.0) * (2.0 * k + 1.0) * PI_D / 64.0) / (2.0 * k + 1.0);
  double w = (2.0 / 16.0) * sin(PI_D * (2.0 * j + 1.0) / 64.0) * s * 0.5;
  w16[j] = (float)w;
}

// analysis table: w[z] * d^l_{mp,m}(beta_z), b=16, layout [l]: 32*off(l) + z*(2l+1)^2 + r
__global__ __launch_bounds__(256) void k_wig16(const float* __restrict__ w16,
                                               float* __restrict__ dw16) {
  int l = blockIdx.y;
  int tw = 2 * l + 1, sq = tw * tw;
  int count = 32 * sq;
  int tid = blockIdx.x * 256 + threadIdx.x;
  if (tid >= count) return;
  int z = tid / sq, r = tid - z * sq;
  int imp = r / tw, im = r - imp * tw;
  double beta = (z + 0.5) * PI_D / 32.0;
  dw16[32 * off_l(l) + tid] = w16[z] * (float)wig_d(l, imp - l, im - l, beta);
}

// synthesis table: (2l+1) * d^l_{mp,m}(beta_z), b=32, layout [l]: 64*off(l) + z*(2l+1)^2 + r
__global__ __launch_bounds__(256) void k_wig32(float* __restrict__ dl32) {
  int l = blockIdx.y;
  int tw = 2 * l + 1, sq = tw * tw;
  int count = 64 * sq;
  int tid = blockIdx.x * 256 + threadIdx.x;
  if (tid >= count) return;
  int z = tid / sq, r = tid - z * sq;
  int imp = r / tw, im = r - imp * tw;
  double beta = (z + 0.5) * PI_D / 64.0;
  dl32[64 * off_l(l) + tid] = (float)((double)tw * wig_d(l, imp - l, im - l, beta));
}

// ---------------------------------------------------------------------------
// Forward: 32x32 DFT2 per (b,c,beta) plane (4096 planes). Small fraction of
// total FLOPs -> plain VALU with fast __sincosf is fine.
// ---------------------------------------------------------------------------
__global__ __launch_bounds__(256) void k_fft2_fwd(const float* __restrict__ x,
                                                  float2* __restrict__ xh) {
  __shared__ float xs[32 * 33];
  __shared__ float Gr[32 * 33], Gi[32 * 33];
  size_t base = (size_t)blockIdx.x * 1024;
  int t = threadIdx.x;
  for (int i = t; i < 1024; i += 256)
    xs[(i >> 5) * 33 + (i & 31)] = x[base + i];
  __syncthreads();
  for (int i = t; i < 1024; i += 256) {      // DFT over gamma (e^{-i})
    int a = i >> 5, n = i & 31;
    float sr = 0.f, si = 0.f;
    for (int g = 0; g < 32; ++g) {
      float th = (float)(2.0 * PI_D / 32.0) * (float)(g * n);
      float s, c; __sincosf(th, &s, &c);
      float v = xs[a * 33 + g];
      sr += v * c; si -= v * s;
    }
    Gr[a * 33 + n] = sr; Gi[a * 33 + n] = si;
  }
  __syncthreads();
  for (int i = t; i < 1024; i += 256) {      // DFT over alpha (e^{-i})
    int p = i >> 5, n = i & 31;
    float sr = 0.f, si = 0.f;
    for (int a = 0; a < 32; ++a) {
      float th = (float)(2.0 * PI_D / 32.0) * (float)(a * p);
      float s, c; __sincosf(th, &s, &c);
      float gr = Gr[a * 33 + n], gi = Gi[a * 33 + n];
      sr += c * gr + s * gi;
      si += c * gi - s * gr;
    }
    float2 o; o.x = sr; o.y = si;
    xh[base + i] = o;
  }
}

// ---------------------------------------------------------------------------
// Wigner analysis: spec[bc][j] = sum_z dw16[l][z][r] * Xh[bc][z][mp%32][m%32]
// ---------------------------------------------------------------------------
__global__ __launch_bounds__(256) void k_analysis(const float2* __restrict__ xh,
                                                  const float* __restrict__ dw16,
                                                  float2* __restrict__ spec) {
  int tid = blockIdx.x * 256 + threadIdx.x;   // exactly 128*5456 threads
  int bc = tid / 5456;
  int j  = tid - bc * 5456;
  int l = 0;
  while (off_l(l + 1) <= j) ++l;
  int r = j - off_l(l);
  int tw = 2 * l + 1;
  int imp = r / tw, im = r - imp * tw;
  int mp = imp - l, m = im - l;
  int pm = mp & 31, pn = m & 31;              // centered mode -> fft bin
  const float2* xp = xh + (size_t)bc * (32 * 1024);
  const float* dp = dw16 + 32 * off_l(l);
  float sr = 0.f, si = 0.f;
  for (int z = 0; z < 32; ++z) {
    float d = dp[z * tw * tw + r];
    float2 v = xp[(size_t)z * 1024 + pm * 32 + pn];
    sr += d * v.x; si += d * v.y;
  }
  float2 o; o.x = sr; o.y = si;
  spec[tid] = o;
}

// ---------------------------------------------------------------------------
// Fused synthesis + 64x64 iDFT2 per (b,c,beta) plane. One WG = one plane.
// Stage 1: build P[64][64] complex in LDS (gather spec through the flat
//          zero-pad map, times (2l+1)d^l; mod-64 scatter is done by owning
//          each output bin and looping l >= max(|m|,|n|)).
// Stage 2: out = Re( E * P * E^T ) / 4096 with E[a][p]=e^{+2pi i a p/64},
//          done as 6 real 64x64x64 GEMMs on v_wmma_f32_16x16x4_f32.
// ---------------------------------------------------------------------------
__global__ __launch_bounds__(256) void k_synth(const float2* __restrict__ spec,
                                               const float* __restrict__ dl32,
                                               float* __restrict__ out) {
  __shared__ float Pr[64 * LSTR], Pi[64 * LSTR];
  __shared__ float Er[64 * LSTR], Ei[64 * LSTR];
  __shared__ float Cr[64 * LSTR], Ci[64 * LSTR];

  int bc = blockIdx.x >> 6;                   // 8192 blocks = bc*64 + z
  int z  = blockIdx.x & 63;
  int t = threadIdx.x;

  // inverse-DFT twiddles e^{+2pi i a p / 64}
  for (int i = t; i < 4096; i += 256) {
    int a = i >> 6, p = i & 63;
    float th = (float)(2.0 * PI_D / 64.0) * (float)(a * p);
    float s, c; __sincosf(th, &s, &c);
    Er[a * LSTR + p] = c; Ei[a * LSTR + p] = s;
  }

  // stage 1: frequency plane P
  const float2* sp = spec + (size_t)bc * 5456;
  for (int i = t; i < 4096; i += 256) {
    int p = i >> 6, q = i & 63;
    float accr = 0.f, acci = 0.f;
    if (p != 32 && q != 32) {                 // Nyquist bins never written
      int m = (p <= 31) ? p : p - 64;
      int n = (q <= 31) ? q : q - 64;
      int am = m < 0 ? -m : m, an = n < 0 ? -n : n;
      int lmin = am > an ? am : an;
      for (int l = lmin; l < 32; ++l) {
        int tw = 2 * l + 1;
        int r = (m + l) * tw + (n + l);
        int j = off_l(l) + r;
        float2 f;
        if (j < 2728)        f = sp[j];                 // flat zero-pad head
        else if (j >= 40952) f = sp[j - 38224];         // flat zero-pad tail
        else                 { f.x = 0.f; f.y = 0.f; }
        float d = dl32[64 * off_l(l) + z * tw * tw + r];
        accr += f.x * d; acci += f.y * d;
      }
    }
    Pr[p * LSTR + q] = accr;
    Pi[p * LSTR + q] = acci;
  }
  __syncthreads();

  int wave = t >> 5, lane = t & 31;
  int lr = lane & 15, lh = lane >> 4;         // half-wave select (K pairs / M+8)

  // GEMM1: C1 = E * P (complex). 32 tile-jobs (4x4 tiles x {re,im}) / 8 waves.
  for (int jj = 0; jj < 4; ++jj) {
    int job = wave * 4 + jj;                  // uniform per wave
    int comp = job & 1;                       // 0 -> Cr, 1 -> Ci
    int tile = job >> 1;
    int tm = (tile >> 2) * 16, tn = (tile & 3) * 16;
    v8f acc = {};
    for (int kt = 0; kt < 16; ++kt) {
      int ks = kt * 4 + lh * 2;
      v2f aR, aI, bR, bI;
      aR.x = Er[(tm + lr) * LSTR + ks];  aR.y = Er[(tm + lr) * LSTR + ks + 1];
      aI.x = Ei[(tm + lr) * LSTR + ks];  aI.y = Ei[(tm + lr) * LSTR + ks + 1];
      bR.x = Pr[ks * LSTR + tn + lr];    bR.y = Pr[(ks + 1) * LSTR + tn + lr];
      bI.x = Pi[ks * LSTR + tn + lr];    bI.y = Pi[(ks + 1) * LSTR + tn + lr];
      if (comp == 0) {                        // Cr = Er*Pr - Ei*Pi
        bI.x = -bI.x; bI.y = -bI.y;
        acc = __builtin_amdgcn_wmma_f32_16x16x4_f32(false, aR, false, bR, (short)0, acc, false, false);
        acc = __builtin_amdgcn_wmma_f32_16x16x4_f32(false, aI, false, bI, (short)0, acc, false, false);
      } else {                                // Ci = Er*Pi + Ei*Pr
        acc = __builtin_amdgcn_wmma_f32_16x16x4_f32(false, aR, false, bI, (short)0, acc, false, false);
        acc = __builtin_amdgcn_wmma_f32_16x16x4_f32(false, aI, false, bR, (short)0, acc, false, false);
      }
    }
    float* Cdst = (comp == 0) ? Cr : Ci;
    for (int r = 0; r < 8; ++r) {
      int row = tm + r + lh * 8;
      Cdst[row * LSTR + tn + lr] = acc[r];
    }
  }
  __syncthreads();

  // GEMM2: out = Cr*Er^T - Ci*Ei^T, scaled by 1/64^2. 16 tiles / 8 waves.
  float* obase = out + ((size_t)bc * 64 + z) * 4096;
  for (int jj = 0; jj < 2; ++jj) {
    int tile = wave * 2 + jj;
    int tm = (tile >> 2) * 16, tn = (tile & 3) * 16;
    v8f acc = {};
    for (int kt = 0; kt < 16; ++kt) {
      int ks = kt * 4 + lh * 2;
      v2f aR, aI, bR, bI;
      aR.x = Cr[(tm + lr) * LSTR + ks];  aR.y = Cr[(tm + lr) * LSTR + ks + 1];
      aI.x = Ci[(tm + lr) * LSTR + ks];  aI.y = Ci[(tm + lr) * LSTR + ks + 1];
      // B = E^T : B[k][n] = E[tn+n][k]
      bR.x =  Er[(tn + lr) * LSTR + ks]; bR.y =  Er[(tn + lr) * LSTR + ks + 1];
      bI.x = -Ei[(tn + lr) * LSTR + ks]; bI.y = -Ei[(tn + lr) * LSTR + ks + 1];
      acc = __builtin_amdgcn_wmma_f32_16x16x4_f32(false, aR, false, bR, (short)0, acc, false, false);
      acc = __builtin_amdgcn_wmma_f32_16x16x4_f32(false, aI, false, bI, (short)0, acc, false, false);
    }
    for (int r = 0; r < 8; ++r) {
      int row = tm + r + lh * 8, col = tn + lr;
      obase[row * 64 + col] = acc[r] * (1.0f / 4096.0f);
    }
  }
}

// ---------------------------------------------------------------------------
// Host side
// ---------------------------------------------------------------------------
extern "C" void kernel_launch(void* const* d_in, const int* in_sizes, int n_in,
                              void* d_out, int out_size, void* d_ws, size_t ws_size,
                              hipStream_t stream) {
  const float* x = (const float*)d_in[0];     // [4,32,32,32,32] f32
  float* out = (float*)d_out;                 // [4,32,64,64,64] f32

  // scratch layout (floats): w16(64 pad) | dw16(174592) | dl32(2795520) |
  //                          xh(4194304 float2) | spec(698368 float2)
  char* ws = (char*)d_ws;
  float*  w16  = (float*)ws;
  float*  dw16 = (float*)(ws + (size_t)64 * 4);
  float*  dl32 = (float*)(ws + (size_t)(64 + 174592) * 4);
  float2* xh   = (float2*)(ws + (size_t)(64 + 174592 + 2795520) * 4);
  float2* spec = (float2*)(ws + (size_t)(64 + 174592 + 2795520) * 4
                              + (size_t)4194304 * sizeof(float2));

  k_weights16<<<1, 32, 0, stream>>>(w16);
  k_wig16<<<dim3(121, 16), 256, 0, stream>>>(w16, dw16);   // 32*31^2 max entries
  k_wig32<<<dim3(993, 32), 256, 0, stream>>>(dl32);        // 64*63^2 max entries
  k_fft2_fwd<<<4096, 256, 0, stream>>>(x, xh);             // B*C*32 planes
  k_analysis<<<2728, 256, 0, stream>>>(xh, dw16, spec);    // 128*5456 threads
  k_synth<<<8192, 256, 0, stream>>>(spec, dl32, out);      // B*C*64 planes
}